// EnhancedTMDO_46686294508074
// MI455X (gfx1250) — compile-verified
//
#include <hip/hip_runtime.h>
#include <hip/hip_bf16.h>
#include <math.h>

typedef __attribute__((ext_vector_type(2))) float v2f;
typedef __attribute__((ext_vector_type(8))) float v8f;

#define SEQ   2048
#define NF    128
#define NB    32
#define ROWS_PER_BLOCK 16
#define STX   130   // LDS row stride for x/h tiles (breaks 16-way bank conflicts on A-frag loads)
#define STG   132   // LDS row stride for GEMM accumulator tile

// ---------------------------------------------------------------------------
// Kernel 1: row softmax of weight[128,128]; store transposed + pair-interleaved
// so B fragments are contiguous 8B loads:
//   wtp[(k>>1)*256 + n*2 + (k&1)] = softmax(weight, axis=1)[n][k]
// ---------------------------------------------------------------------------
__global__ void tmdo_softmax_kernel(const float* __restrict__ w,
                                    float* __restrict__ wtp) {
    __shared__ float red[NF];
    const int g = blockIdx.x;     // row of w
    const int t = threadIdx.x;    // column index k
    float v = w[g * NF + t];

    red[t] = v;
    __syncthreads();
    for (int s = 64; s > 0; s >>= 1) {
        if (t < s) red[t] = fmaxf(red[t], red[t + s]);
        __syncthreads();
    }
    const float mx = red[0];
    __syncthreads();

    const float e = expf(v - mx);
    red[t] = e;
    __syncthreads();
    for (int s = 64; s > 0; s >>= 1) {
        if (t < s) red[t] += red[t + s];
        __syncthreads();
    }
    const float p = e / red[0];

    wtp[(t >> 1) * 256 + g * 2 + (t & 1)] = p;
}

// ---------------------------------------------------------------------------
// Kernel 2: per block = 16 flattened rows x 128 features.
//   indicator GEMM via v_wmma_f32_16x16x4_f32 (8 waves, one 16-col N-tile each,
//   32 K-steps), plus time-diff and separable 3x3 Laplacian from an LDS-staged
//   18-row halo tile.
// ---------------------------------------------------------------------------
__global__ __launch_bounds__(256)
void tmdo_main_kernel(const float* __restrict__ x,
                      const float* __restrict__ wtp,
                      float* __restrict__ out_tmdo,
                      float* __restrict__ out_lap) {
    __shared__ float smem_x[18 * STX];               // rows s0-1 .. s0+16
    __shared__ float smem_h[18 * STX];               // horizontal [1,-2,1] filter of x
    __shared__ float smem_g[ROWS_PER_BLOCK * STG];   // GEMM result tile

    const int tid  = threadIdx.x;
    const int m0   = blockIdx.x * ROWS_PER_BLOCK;    // first flattened row
    const int sl0  = m0 & (SEQ - 1);                 // row-in-batch (tile never crosses batch)
    const size_t batch_base = (size_t)(m0 - sl0) * NF;

    // ---- stage x tile (with zero-filled batch-edge halos) ----
    for (int idx = tid; idx < 18 * NF; idx += 256) {
        const int i = idx >> 7;          // 0..17  -> s_local = sl0 - 1 + i
        const int f = idx & (NF - 1);
        const int sl = sl0 - 1 + i;
        float v = 0.0f;
        if (sl >= 0 && sl < SEQ) v = x[batch_base + (size_t)sl * NF + f];
        smem_x[i * STX + f] = v;
    }
    __syncthreads();

    // ---- horizontal [1,-2,1] pass for the separable Laplacian ----
    for (int idx = tid; idx < 18 * NF; idx += 256) {
        const int i = idx >> 7;
        const int f = idx & (NF - 1);
        const float xc = smem_x[i * STX + f];
        const float xl = (f > 0)      ? smem_x[i * STX + f - 1] : 0.0f;
        const float xr = (f < NF - 1) ? smem_x[i * STX + f + 1] : 0.0f;
        smem_h[i * STX + f] = xl - 2.0f * xc + xr;
    }

    // ---- indicator GEMM: D = A(16x128) * B(128x16) per wave ----
    // A layout (f32 16x4): lane l -> M = l&15, kpair base = 2*(l>>4)
    // B layout (f32 4x16): lane l -> N = l&15, kpair base = 2*(l>>4)
    {
        const int lane  = tid & 31;
        const int wave  = tid >> 5;          // 0..7 -> N-tile
        const int mrow  = lane & 15;
        const int khalf = lane >> 4;         // 0 or 1
        const int n_abs = wave * 16 + mrow;  // global column (feature g)
        const float* arow = &smem_x[(mrow + 1) * STX];  // center rows are i = M+1

        v8f acc = {};
        #pragma unroll
        for (int ks = 0; ks < 32; ++ks) {
            const int kb = ks * 4 + khalf * 2;
            const v2f a = *(const v2f*)(arow + kb);                 // ds_load_b64
            const v2f b = *(const v2f*)(wtp + kb * NF + 2 * n_abs); // global_load_b64 (L2-hot)
            acc = __builtin_amdgcn_wmma_f32_16x16x4_f32(
                      false, a, false, b, (short)0, acc, false, false);
        }
        // C/D layout: VGPR r, lane l -> (M = r + 8*(l>>4), N = l&15)
        #pragma unroll
        for (int r = 0; r < 8; ++r) {
            smem_g[(r + khalf * 8) * STG + n_abs] = acc[r];
        }
    }
    __syncthreads();

    // ---- epilogue: tmdo = 0.5*time_diff + 0.5*(x - gemm); lap = vertical pass ----
    for (int idx = tid; idx < ROWS_PER_BLOCK * NF; idx += 256) {
        const int M = idx >> 7;
        const int f = idx & (NF - 1);
        const int i = M + 1;
        const int sl = sl0 + M;

        const float xc = smem_x[i * STX + f];

        float td = 0.0f;
        if (sl >= 1 && sl <= SEQ - 2) {
            td = smem_x[(i + 1) * STX + f] - 2.0f * xc + smem_x[(i - 1) * STX + f];
        }
        const float ind  = xc - smem_g[M * STG + f];
        const float tmdo = 0.5f * td + 0.5f * ind;

        const float lap = smem_h[(i - 1) * STX + f]
                        - 2.0f * smem_h[i * STX + f]
                        + smem_h[(i + 1) * STX + f];

        const size_t o = (size_t)(m0 + M) * NF + f;
        out_tmdo[o] = tmdo;
        out_lap[o]  = lap;
    }
}

// ---------------------------------------------------------------------------
extern "C" void kernel_launch(void* const* d_in, const int* in_sizes, int n_in,
                              void* d_out, int out_size, void* d_ws, size_t ws_size,
                              hipStream_t stream) {
    (void)n_in; (void)ws_size; (void)out_size;
    const float* x = (const float*)d_in[0];          // [32,2048,128] f32
    const float* w = (const float*)d_in[1];          // [128,128] f32
    float* wtp = (float*)d_ws;                       // 64 KB interleaved softmax(w).T

    const int total_rows = in_sizes[0] / NF;         // 65536
    float* out_tmdo = (float*)d_out;
    float* out_lap  = out_tmdo + (size_t)total_rows * NF;

    tmdo_softmax_kernel<<<NF, NF, 0, stream>>>(w, wtp);
    tmdo_main_kernel<<<total_rows / ROWS_PER_BLOCK, 256, 0, stream>>>(
        x, wtp, out_tmdo, out_lap);
}